// MLPModel_23124103921976
// MI455X (gfx1250) — compile-verified
//
#include <hip/hip_runtime.h>
#include <hip/hip_bf16.h>

// Problem constants (from reference): B=8192, R=64, V=32, H=128
#define BB 8192
#define RR 64
#define VV 32
#define HH 128
#define OUT_ROW (RR * VV + VV * VV)   // 2048 + 1024 = 3072 floats per batch row
#define ROWS_PER_BLOCK 16

typedef float v2f __attribute__((ext_vector_type(2)));
typedef float v8f __attribute__((ext_vector_type(8)));
typedef float f4  __attribute__((ext_vector_type(4)));

__device__ __forceinline__ float sigmoidf(float x) {
    return 1.0f / (1.0f + __expf(-x));
}

// ---------------------------------------------------------------------------
// Kernel 1: build the two 32x32 lookup tables in workspace.
//   Block 0 (128 thr): S = sigmoid(W1+b1) (32x128, LDS), then
//                      NodeTable = S @ W2 + b2 via V_WMMA_F32_16X16X4_F32
//                      (4 waves, one 16x16 tile each, K-loop of 32 WMMAs).
//   Blocks 1..8 (128 thr each): 1024 pair entries, one 128-length dot each.
// Workspace layout: tabs[0..1023] = NodeTable (row-major 32x32)
//                   tabs[1024..2047] = PairTable flat (i*32+j)
// ---------------------------------------------------------------------------
__global__ void mlp_tables_kernel(const float* __restrict__ W1,
                                  const float* __restrict__ b1,
                                  const float* __restrict__ W2,
                                  const float* __restrict__ b2,
                                  const float* __restrict__ Wp1,
                                  const float* __restrict__ bp1,
                                  const float* __restrict__ Wp2,
                                  const float* __restrict__ bp2,
                                  float* __restrict__ tabs) {
    if (blockIdx.x == 0) {
        // ---- node table via f32 WMMA ----
        __shared__ float S[VV * HH];                 // 16 KB
        int t = threadIdx.x;                          // 128 threads
        for (int e = t; e < VV * HH; e += 128) {
            int h = e & (HH - 1);
            S[e] = sigmoidf(W1[e] + b1[h]);           // W1 row-major (V,H)
        }
        __syncthreads();

        int wave = t >> 5;
        int lane = t & 31;
        if (wave < 4) {
            // tile origin for this wave
            const int m0 = (wave >> 1) << 4;
            const int n0 = (wave & 1) << 4;
            const int l  = lane & 15;
            const int khalf = (lane >> 4) << 1;       // lanes 0-15 -> K+0/1, 16-31 -> K+2/3

            v8f c = {};
            #pragma unroll 4
            for (int k = 0; k < HH; k += 4) {
                int kk = k + khalf;
                // A fragment: 16x4 f32. lane L<16: v0=A[L][k],  v1=A[L][k+1]
                //                       lane L>=16: v0=A[L][k+2],v1=A[L][k+3]
                v2f a;
                a.x = S[(m0 + l) * HH + kk];
                a.y = S[(m0 + l) * HH + kk + 1];
                // B fragment: 4x16 f32, rows striped across lanes per VGPR,
                // mirroring the A-side K split (low half rows lanes 0-15).
                v2f bm;
                bm.x = W2[(kk)     * VV + n0 + l];
                bm.y = W2[(kk + 1) * VV + n0 + l];
                c = __builtin_amdgcn_wmma_f32_16x16x4_f32(
                        /*neg_a=*/false, a, /*neg_b=*/false, bm,
                        /*c_mod=*/(short)0, c, /*reuse_a=*/false, /*reuse_b=*/false);
            }
            // C/D layout: VGPR g -> lanes 0-15: M=g, N=lane; lanes 16-31: M=g+8
            const int col   = n0 + l;
            const int rbase = m0 + ((lane >> 4) << 3);
            const float bias = b2[col];
            #pragma unroll
            for (int g = 0; g < 8; ++g) {
                tabs[(rbase + g) * VV + col] = c[g] + bias;
            }
        }
    } else {
        // ---- pair table: 8 blocks x 128 threads = 1024 entries ----
        int p = (blockIdx.x - 1) * 128 + threadIdx.x;     // 0..1023
        int i = p >> 5;
        int j = p & 31;
        const float* wa = Wp1 + i * HH;
        const float* wb = Wp1 + (VV + j) * HH;
        float acc = 0.0f;
        #pragma unroll 8
        for (int h = 0; h < HH; ++h) {
            acc += sigmoidf(wa[h] + wb[h] + bp1[h]) * Wp2[h];
        }
        tabs[VV * VV + p] = acc + bp2[0];
    }
}

// ---------------------------------------------------------------------------
// Kernel 2: bandwidth kernel. Stage the 8KB of tables in LDS, then for each
// batch row emit 768 float4 stores: node part = LDS gather by obs index,
// pair part = LDS broadcast. Stores are coalesced, 128-bit, non-temporal
// (96 MiB streaming output, zero reuse).
// ---------------------------------------------------------------------------
__global__ void mlp_broadcast_kernel(const int* __restrict__ obs,
                                     const float* __restrict__ tabs,
                                     float* __restrict__ out) {
    __shared__ f4 nodeT[VV * (VV / 4)];        // 32 rows x 8 float4 = 4 KB
    __shared__ f4 pairT[(VV * VV) / 4];        // 256 float4 = 4 KB
    __shared__ int obs_s[ROWS_PER_BLOCK * RR]; // 1024 ints = 4 KB

    const int t = threadIdx.x;                 // 256 threads (8 waves)
    const f4* tab4 = (const f4*)tabs;
    nodeT[t] = tab4[t];
    pairT[t] = tab4[256 + t];

    const int base_b = blockIdx.x * ROWS_PER_BLOCK;
    const int* obs_blk = obs + (size_t)base_b * RR;
    #pragma unroll
    for (int e = t; e < ROWS_PER_BLOCK * RR; e += 256) obs_s[e] = obs_blk[e];
    __syncthreads();

    for (int rb = 0; rb < ROWS_PER_BLOCK; ++rb) {
        const size_t b = (size_t)base_b + rb;
        f4* orow = (f4*)(out + b * OUT_ROW);
        #pragma unroll
        for (int jj = 0; jj < 3; ++jj) {       // 3*256 = 768 float4 per row
            const int i = t + jj * 256;
            f4 v;
            if (i < 512) {                      // node chunk: [r=i>>3][cols 4*(i&7)..]
                const int idx = obs_s[rb * RR + (i >> 3)];
                v = nodeT[idx * 8 + (i & 7)];
            } else {                            // pair broadcast chunk
                v = pairT[i - 512];
            }
            __builtin_nontemporal_store(v, orow + i);
        }
    }
}

extern "C" void kernel_launch(void* const* d_in, const int* in_sizes, int n_in,
                              void* d_out, int out_size, void* d_ws, size_t ws_size,
                              hipStream_t stream) {
    const int*   obs = (const int*)  d_in[0];
    const float* W1  = (const float*)d_in[1];
    const float* b1  = (const float*)d_in[2];
    const float* W2  = (const float*)d_in[3];
    const float* b2  = (const float*)d_in[4];
    const float* Wp1 = (const float*)d_in[5];
    const float* bp1 = (const float*)d_in[6];
    const float* Wp2 = (const float*)d_in[7];
    const float* bp2 = (const float*)d_in[8];
    float* tabs = (float*)d_ws;                // needs 2048 floats = 8 KB
    float* out  = (float*)d_out;

    mlp_tables_kernel<<<9, 128, 0, stream>>>(W1, b1, W2, b2, Wp1, bp1, Wp2, bp2, tabs);
    mlp_broadcast_kernel<<<BB / ROWS_PER_BLOCK, 256, 0, stream>>>(obs, tabs, out);
}